// TargetAnalyticalTC_33414845563116
// MI455X (gfx1250) — compile-verified
//
#include <hip/hip_runtime.h>

// einsum 'pbcik,bckj->pbcij' with M=K=N=3, P=4, B=8192, C=128, fp32.
// Memory-bound (0.66 FLOP/byte, ~340 MB total => ~14.6us at 23.3 TB/s).
// Design (final):
//  - T2 tile (36 KB per 256-thread block) staged into LDS with CDNA5 async
//    copies (global_load_async_to_lds_b128, ASYNCcnt). One shared address
//    register pair + literal offset: immediates (INST_OFFSET applies to both
//    the LDS and global addresses per the ISA), fully coalesced 16B stride.
//  - T2 read back conflict-free (stride 36 DWORDs = 9 x 16B) as ds_load_b128.
//  - T1 and Out are single-use streams: direct non-temporal b128 accesses,
//    with global_prefetch_b8 priming the next 37.7 MB-strided T1 slice.
//  - WMMA intentionally absent: fp32 WMMA is 16x16x4 only; block-diagonal
//    packing of 3x3 problems is 45/256 useful with per-lane scatter -- pure
//    overhead in a kernel whose VALU is <15% busy at the HBM floor.

typedef float v4f __attribute__((ext_vector_type(4)));

#define MK        9                       // 3x3 floats per matrix
#define PAIRS     4                       // (b,c) pairs per thread (144B tiles)
#define P_DIM     4
#define BLOCK     256
#define TILE_PAIRS   (BLOCK * PAIRS)      // 1024 pairs per block
#define TILE_FLOATS  (TILE_PAIRS * MK)    // 9216 floats = 36 KB

__global__ __launch_bounds__(BLOCK)
void bmm3x3_async(const float* __restrict__ T1,
                  const float* __restrict__ T2,
                  float* __restrict__ Out,
                  long long pstride_f)    // floats between p-slices = B*C*9
{
    __shared__ float ldsT2[TILE_FLOATS];

    const int       tid       = threadIdx.x;
    const long long tileBaseF = (long long)blockIdx.x * TILE_FLOATS;
    const long long base      = tileBaseF + (long long)tid * (PAIRS * MK);

    // ---- cooperative async fill of the block's T2 tile (CDNA5 async engine).
    // 2304 x b128 transfers per block, 16B lane stride: fully coalesced.
    // One VGPR pair + one LDS-offset VGPR; the 9 slices use offset: imms.
    {
        const unsigned ldsOff = (unsigned)(unsigned long long)(&ldsT2[0]) + tid * 16u;
        const float*   gp     = T2 + tileBaseF + tid * 4;
#define ASYNC_CP(OFF)                                                         \
        asm volatile("global_load_async_to_lds_b128 %0, %1, off offset:" #OFF \
                     :: "v"(ldsOff), "v"(gp) : "memory")
        ASYNC_CP(0);
        ASYNC_CP(4096);
        ASYNC_CP(8192);
        ASYNC_CP(12288);
        ASYNC_CP(16384);
        ASYNC_CP(20480);
        ASYNC_CP(24576);
        ASYNC_CP(28672);
        ASYNC_CP(32768);
#undef ASYNC_CP
        asm volatile("s_wait_asynccnt 0" ::: "memory");
    }
    __syncthreads();

    // ---- pull this thread's 4 T2 blocks from LDS (16B-aligned, stride 144B
    //      => 36-DWORD lane stride, bank-conflict-free) into registers.
    float t2s[PAIRS * MK];
    {
        const v4f* l2 = (const v4f*)(ldsT2 + tid * (PAIRS * MK));
        #pragma unroll
        for (int v = 0; v < 9; ++v) {
            v4f x = l2[v];
            #pragma unroll
            for (int c = 0; c < 4; ++c) t2s[4 * v + c] = x[c];
        }
    }

    #pragma unroll
    for (int p = 0; p < P_DIM; ++p) {
        const float* t1base = T1 + base + (long long)p * pstride_f;

        // Speculative prefetch of the next far-strided T1 slice.
        __builtin_prefetch(t1base + pstride_f, 0, 1);

        // ---- T1 tile for this p: 36 floats, single-use -> NT b128 loads
        float t1s[PAIRS * MK];
        {
            const v4f* p1 = (const v4f*)t1base;
            #pragma unroll
            for (int v = 0; v < 9; ++v) {
                v4f x = __builtin_nontemporal_load(p1 + v);
                #pragma unroll
                for (int c = 0; c < 4; ++c) t1s[4 * v + c] = x[c];
            }
        }

        // ---- 4 independent 3x3 x 3x3 matmuls, fully unrolled (108 FMAs)
        float os[PAIRS * MK];
        #pragma unroll
        for (int q = 0; q < PAIRS; ++q) {
            #pragma unroll
            for (int i = 0; i < 3; ++i) {
                #pragma unroll
                for (int j = 0; j < 3; ++j) {
                    os[q * 9 + i * 3 + j] =
                        fmaf(t1s[q * 9 + i * 3 + 0], t2s[q * 9 + 0 + j],
                        fmaf(t1s[q * 9 + i * 3 + 1], t2s[q * 9 + 3 + j],
                             t1s[q * 9 + i * 3 + 2] * t2s[q * 9 + 6 + j]));
                }
            }
        }

        // ---- output tile: 36 floats, single-use -> NT b128 stores
        {
            v4f* po = (v4f*)(Out + base + (long long)p * pstride_f);
            #pragma unroll
            for (int v = 0; v < 9; ++v) {
                v4f x;
                #pragma unroll
                for (int c = 0; c < 4; ++c) x[c] = os[4 * v + c];
                __builtin_nontemporal_store(x, po + v);
            }
        }
    }
}

extern "C" void kernel_launch(void* const* d_in, const int* in_sizes, int n_in,
                              void* d_out, int out_size, void* d_ws, size_t ws_size,
                              hipStream_t stream) {
    const float* T1  = (const float*)d_in[0];   // (P, B, C, 3, 3) fp32
    const float* T2  = (const float*)d_in[1];   // (1, B, C, 3, 3) fp32
    float*       Out = (float*)d_out;           // (P, B, C, 3, 3) fp32

    const long long pstride_f = in_sizes[1];                 // B*C*9 per p-slice
    const long long nblocks   = pstride_f / (long long)TILE_FLOATS; // 1024

    dim3 blk(BLOCK);
    dim3 grd((unsigned)nblocks);
    hipLaunchKernelGGL(bmm3x3_async, grd, blk, 0, stream, T1, T2, Out, pstride_f);
}